// SimpleBackFlow_12429635355196
// MI455X (gfx1250) — compile-verified
//
#include <hip/hip_runtime.h>

typedef __attribute__((ext_vector_type(2))) float v2f;
typedef __attribute__((ext_vector_type(8))) float v8f;

// Branchless tanh: prefer native V_TANH on gfx1250, else exp2/rcp formulation
// tanh(u) = 1 - 2/(exp(2u)+1); exp(2u) = exp2(u * 2/ln2).
// Limits: u->+inf: exp2->inf, rcp->0, t->1.  u->-inf: exp2->0, t->-1.
__device__ __forceinline__ float tanh_fast(float u) {
#if __has_builtin(__builtin_amdgcn_tanhf)
    return __builtin_amdgcn_tanhf(u);
#else
    float e = __builtin_amdgcn_exp2f(u * 2.8853900817779268f); // 2*log2(e)
    return 1.0f - 2.0f * __builtin_amdgcn_rcpf(e + 1.0f);
#endif
}

// jax.nn.gelu default (approximate=True): 0.5x(1+tanh(sqrt(2/pi)(x+0.044715x^3)))
__device__ __forceinline__ float gelu_t(float x) {
    float u = 0.7978845608028654f * (x + 0.044715f * x * x * x);
    return 0.5f * x * (1.0f + tanh_fast(u));
}

// One wave per (b,i). N=64 neighbors processed as 4 tiles of 16 columns.
// Transposed formulation: weights (W^T, padded to 16 rows) in A operand,
// data columns in B operand; layer-3 WMMA chain-accumulates the neighbor sum.
__global__ __launch_bounds__(256) void backflow_wmma_kernel(
    const float* __restrict__ dist,   // [B,N,N]
    const float* __restrict__ diff,   // [B,N,N,3]
    const float* __restrict__ W1,     // [4,4]
    const float* __restrict__ b1,     // [4]
    const float* __restrict__ W2,     // [4,4]
    const float* __restrict__ b2,     // [4]
    const float* __restrict__ Wp,     // [4,3]
    const float* __restrict__ shift,  // [3]
    float* __restrict__ out,          // [B,N,3]
    int npairs)                       // B*N
{
    const int wave = (int)((blockIdx.x * blockDim.x + threadIdx.x) >> 5);
    if (wave >= npairs) return;                  // uniform per wave: EXEC stays full
    const int lane = (int)(threadIdx.x & 31);
    const int nl   = lane & 15;                  // column / M-row slot within half
    const bool hi  = lane >= 16;
    const int k0   = hi ? 2 : 0;                 // K indices striped across lane halves
    const int k1   = hi ? 3 : 1;
    const int i    = wave & 63;                  // node index within N=64
    const size_t rowbase = (size_t)wave * 64;    // flat pair-row base = (b*64 + i)*64

    // ---- A operands: W^T padded to 16x4 (lane = M row, v.x = K k0, v.y = K k1) ----
    v2f A1 = {0.f, 0.f}, A2 = {0.f, 0.f}, A3 = {0.f, 0.f};
    if (nl < 4) {
        A1.x = W1[k0 * 4 + nl];  A1.y = W1[k1 * 4 + nl];   // W1^T[m][k] = W1[k][m]
        A2.x = W2[k0 * 4 + nl];  A2.y = W2[k1 * 4 + nl];
    }
    if (nl < 3) {
        A3.x = Wp[k0 * 3 + nl];  A3.y = Wp[k1 * 3 + nl];   // Wproj^T[s][k] = Wp[k][s]
    }

    // Per-lane bias vectors in B layout (row k0 in low lanes, k1 in high pattern)
    const float bias1_0 = b1[k0], bias1_1 = b1[k1];
    const float bias2_0 = b2[k0], bias2_1 = b2[k1];

    const int diagTile = i >> 4;
    const int diagCol  = i & 15;

    v8f acc = {};        // contr^T accumulator (rows s=0..2 valid, cols = lanes 0-15)
    v8f zero = {};

#pragma unroll
    for (int t = 0; t < 4; ++t) {
        // ---- B operand: X^T tile, 4x16. lane n: v.x = feat k0, v.y = feat k1 ----
        const size_t row = rowbase + (size_t)(t * 16 + nl);
        const float* p0 = hi ? (diff + row * 3 + 1) : (dist + row);
        const float* p1 = hi ? (diff + row * 3 + 2) : (diff + row * 3);
        v2f bx;
        bx.x = *p0;
        bx.y = *p1;

        // ---- layer 1: H^T = W1^T @ X^T ----
        v8f h = __builtin_amdgcn_wmma_f32_16x16x4_f32(
            false, A1, false, bx, (short)0, zero, false, false);

        // D -> B layout: rows 2,3 (low lanes of v2,v3) move to high lanes
        float h2 = __shfl_xor(h[2], 16, 32);
        float h3 = __shfl_xor(h[3], 16, 32);
        v2f bh;
        bh.x = gelu_t((hi ? h2 : h[0]) + bias1_0);
        bh.y = gelu_t((hi ? h3 : h[1]) + bias1_1);

        // ---- layer 2: E^T = W2^T @ gelu(H^T + b1) ----
        v8f e = __builtin_amdgcn_wmma_f32_16x16x4_f32(
            false, A2, false, bh, (short)0, zero, false, false);

        float e2 = __shfl_xor(e[2], 16, 32);
        float e3 = __shfl_xor(e[3], 16, 32);
        v2f be;
        be.x = gelu_t((hi ? e2 : e[0]) + bias2_0);
        be.y = gelu_t((hi ? e3 : e[1]) + bias2_1);

        // remove_diag: zero the diagonal column (j == i) so it contributes 0
        if (t == diagTile) {
            const bool isdiag = (nl == diagCol);
            be.x = isdiag ? 0.f : be.x;
            be.y = isdiag ? 0.f : be.y;
        }

        // ---- layer 3 with chained accumulation over tiles ----
        acc = __builtin_amdgcn_wmma_f32_16x16x4_f32(
            false, A3, false, be, (short)0, acc, false, false);
    }

    // Reduce over the 16 neighbor columns (lanes 0-15) for s = 0,1,2
    float r0 = acc[0], r1 = acc[1], r2 = acc[2];
#pragma unroll
    for (int m = 1; m <= 8; m <<= 1) {
        r0 += __shfl_xor(r0, m, 32);
        r1 += __shfl_xor(r1, m, 32);
        r2 += __shfl_xor(r2, m, 32);
    }

    if (lane == 0) {
        float* o = out + (size_t)wave * 3;
        o[0] = r0 + shift[0];
        o[1] = r1 + shift[1];
        o[2] = r2 + shift[2];
    }
}

extern "C" void kernel_launch(void* const* d_in, const int* in_sizes, int n_in,
                              void* d_out, int out_size, void* d_ws, size_t ws_size,
                              hipStream_t stream) {
    // setup_inputs order: ri, rij_dist, rij_diff, W1, b1, W2, b2, Wproj, shift
    const float* dist  = (const float*)d_in[1];
    const float* diff  = (const float*)d_in[2];
    const float* W1    = (const float*)d_in[3];
    const float* b1    = (const float*)d_in[4];
    const float* W2    = (const float*)d_in[5];
    const float* b2    = (const float*)d_in[6];
    const float* Wp    = (const float*)d_in[7];
    const float* shift = (const float*)d_in[8];
    float* out = (float*)d_out;

    const int npairs  = in_sizes[0] / 3;     // ri is [B,N,3] -> B*N
    const int threads = 256;                 // 8 waves per block
    const int blocks  = (npairs * 32 + threads - 1) / threads;

    hipLaunchKernelGGL(backflow_wmma_kernel, dim3(blocks), dim3(threads), 0, stream,
                       dist, diff, W1, b1, W2, b2, Wp, shift, out, npairs);
}